// modeler_55800215110110
// MI455X (gfx1250) — compile-verified
//
#include <hip/hip_runtime.h>
#include <hip/hip_bf16.h>

typedef float v2f __attribute__((ext_vector_type(2)));
typedef float v8f __attribute__((ext_vector_type(8)));

#define NPb   8000
#define FTb   512
#define OUTb  256

// epilogue modes
#define MODE_PLAIN    0   // x = alpha*acc (+Cadd) (+bias)
#define MODE_RELU     1   // x = relu(acc)
#define MODE_AVG_RELU 2   // x = 0.5*(relu(acc) + Cadd)

// ---------------------------------------------------------------------------
// Tiled fp32 WMMA GEMM:  C(MxN) = epilogue( A(MxK) @ B(KxN) )
// A,B,C row-major. K multiple of 16, N multiple of 128, M guarded.
// Block = 256 threads (8 waves). Tile 128x128x16. Wave tile 32(M)x64(N).
// ---------------------------------------------------------------------------
#define LDA_S 18   // LDS A row stride ([128][18], row=m col=k) -> conflict-free b64 frag reads
#define LDB_S 136  // LDS B row stride ([16][136], row=k col=n) -> conflict-free frag reads & b128 fills

__device__ __forceinline__ void load_tile(const float* __restrict__ A,
                                          const float* __restrict__ B,
                                          int M, int lda, int ldb,
                                          int m_blk, int n_blk, int kt,
                                          int rA0, int cA, int rB0, int cB,
                                          float4& a0, float4& a1, float4& b0, float4& b1)
{
    const int k0 = kt * 16;
    const int gm0 = m_blk + rA0;
    const int gm1 = gm0 + 64;
    const float4 z = make_float4(0.f, 0.f, 0.f, 0.f);
    a0 = (gm0 < M) ? *reinterpret_cast<const float4*>(A + (size_t)gm0 * lda + k0 + cA) : z;
    a1 = (gm1 < M) ? *reinterpret_cast<const float4*>(A + (size_t)gm1 * lda + k0 + cA) : z;
    b0 = *reinterpret_cast<const float4*>(B + (size_t)(k0 + rB0) * ldb + n_blk + cB);
    b1 = *reinterpret_cast<const float4*>(B + (size_t)(k0 + rB0 + 8) * ldb + n_blk + cB);
}

__device__ __forceinline__ void store_tile(float* sAb, float* sBb,
                                           int rA0, int cA, int rB0, int cB,
                                           const float4& a0, const float4& a1,
                                           const float4& b0, const float4& b1)
{
    float* pa0 = &sAb[rA0 * LDA_S + cA];
    pa0[0] = a0.x; pa0[1] = a0.y; pa0[2] = a0.z; pa0[3] = a0.w;
    float* pa1 = &sAb[(rA0 + 64) * LDA_S + cA];
    pa1[0] = a1.x; pa1[1] = a1.y; pa1[2] = a1.z; pa1[3] = a1.w;
    *reinterpret_cast<float4*>(&sBb[rB0 * LDB_S + cB])       = b0;
    *reinterpret_cast<float4*>(&sBb[(rB0 + 8) * LDB_S + cB]) = b1;
}

__device__ __forceinline__ void compute_tile(const float* sAb, const float* sBb,
                                             int wm, int wn, int lmod, int lhalf,
                                             v8f acc[2][4])
{
#pragma unroll
    for (int ks = 0; ks < 4; ++ks) {
        const int kk = ks * 4 + lhalf * 2;    // frag K index: 2*(lane/16)+v (+4*ks)
        v2f af[2];
#pragma unroll
        for (int mt = 0; mt < 2; ++mt)        // A frag: lane holds row M=lane%16
            af[mt] = *reinterpret_cast<const v2f*>(&sAb[(wm + mt * 16 + lmod) * LDA_S + kk]);
        v2f bf[4];
#pragma unroll
        for (int nt = 0; nt < 4; ++nt) {      // B frag: lane holds col N=lane%16
            const int n = wn + nt * 16 + lmod;
            v2f b;
            b.x = sBb[kk * LDB_S + n];
            b.y = sBb[(kk + 1) * LDB_S + n];
            bf[nt] = b;
        }
#pragma unroll
        for (int mt = 0; mt < 2; ++mt)
#pragma unroll
            for (int nt = 0; nt < 4; ++nt)
                acc[mt][nt] = __builtin_amdgcn_wmma_f32_16x16x4_f32(
                    false, af[mt], false, bf[nt], (short)0, acc[mt][nt], false, false);
    }
}

__global__ void __launch_bounds__(256) gemm128(
    const float* __restrict__ A, const float* __restrict__ B,
    float* __restrict__ C, const float* __restrict__ Cadd,
    const float* __restrict__ bias,
    int M, int N, int K, int lda, int ldb, int ldc,
    float alpha, int mode)
{
    __shared__ float sA[2][128 * LDA_S];
    __shared__ float sB[2][16 * LDB_S];

    const int t     = threadIdx.x;
    const int lane  = t & 31;
    const int wave  = t >> 5;
    const int lmod  = lane & 15;
    const int lhalf = lane >> 4;
    const int m_blk = blockIdx.x * 128;
    const int n_blk = blockIdx.y * 128;
    const int wm = (wave & 3) * 32;   // 4 waves along M
    const int wn = (wave >> 2) * 64;  // 2 waves along N
    // global->LDS fill assignments
    const int rA0 = t >> 2;           // A rows t/4 and t/4+64, 4 floats each
    const int cA  = (t & 3) * 4;
    const int rB0 = t >> 5;           // B rows t/32 and t/32+8, 4 floats each
    const int cB  = (t & 31) * 4;

    v8f acc[2][4];
#pragma unroll
    for (int i = 0; i < 2; ++i)
#pragma unroll
        for (int j = 0; j < 4; ++j)
#pragma unroll
            for (int e = 0; e < 8; ++e) acc[i][j][e] = 0.0f;

    const int ktiles = K >> 4;
    float4 a0, a1, b0, b1;
    load_tile(A, B, M, lda, ldb, m_blk, n_blk, 0, rA0, cA, rB0, cB, a0, a1, b0, b1);
    store_tile(&sA[0][0], &sB[0][0], rA0, cA, rB0, cB, a0, a1, b0, b1);
    __syncthreads();

    int buf = 0;
    for (int kt = 0; kt < ktiles; ++kt) {
        const bool next = (kt + 1 < ktiles);
        if (next)
            load_tile(A, B, M, lda, ldb, m_blk, n_blk, kt + 1, rA0, cA, rB0, cB, a0, a1, b0, b1);
        if (kt + 2 < ktiles) {  // -> global_prefetch_b8
            const int gm0 = m_blk + rA0;
            if (gm0 < M) __builtin_prefetch(A + (size_t)gm0 * lda + (kt + 2) * 16 + cA, 0, 1);
            __builtin_prefetch(B + (size_t)((kt + 2) * 16 + rB0) * ldb + n_blk + cB, 0, 1);
        }
        compute_tile(&sA[buf][0], &sB[buf][0], wm, wn, lmod, lhalf, acc);
        if (next)
            store_tile(&sA[buf ^ 1][0], &sB[buf ^ 1][0], rA0, cA, rB0, cB, a0, a1, b0, b1);
        __syncthreads();
        buf ^= 1;
    }

    // epilogue: C/D layout -> M = v + 8*(lane/16), N = lane%16
#pragma unroll
    for (int mt = 0; mt < 2; ++mt) {
#pragma unroll
        for (int nt = 0; nt < 4; ++nt) {
            const int n  = n_blk + wn + nt * 16 + lmod;
            const int mb = m_blk + wm + mt * 16 + lhalf * 8;
#pragma unroll
            for (int v = 0; v < 8; ++v) {
                const int m = mb + v;
                if (m < M) {
                    const size_t idx = (size_t)m * ldc + n;
                    float x = acc[mt][nt][v];
                    if (mode == MODE_RELU) {
                        x = fmaxf(x, 0.0f);
                    } else if (mode == MODE_AVG_RELU) {
                        x = 0.5f * (fmaxf(x, 0.0f) + Cadd[idx]);
                    } else {
                        x *= alpha;
                        if (Cadd) x += Cadd[idx];
                        if (bias) x += bias[n];
                    }
                    C[idx] = x;
                }
            }
        }
    }
}

// ---------------------------------------------------------------------------
// C(256x256) = A^T @ B, with A and B stored (K x 256) row-major.
// One wave per 16x16 output tile; streams K from L2 (operands are L2-resident).
// ---------------------------------------------------------------------------
__global__ void __launch_bounds__(32) gemm_atb256(
    const float* __restrict__ A, const float* __restrict__ B,
    float* __restrict__ C, int K)
{
    const int lane  = threadIdx.x & 31;
    const int lmod  = lane & 15;
    const int lhalf = lane >> 4;
    const int m0 = blockIdx.x * 16;
    const int n0 = blockIdx.y * 16;

    v8f acc;
#pragma unroll
    for (int e = 0; e < 8; ++e) acc[e] = 0.0f;

    const float* Ap = A + (size_t)(lhalf * 2) * 256 + m0 + lmod;
    const float* Bp = B + (size_t)(lhalf * 2) * 256 + n0 + lmod;
    for (int k = 0; k < K; k += 16) {
#pragma unroll
        for (int u = 0; u < 4; ++u) {
            const size_t o = (size_t)(k + u * 4) * 256;
            v2f a, b;
            a.x = Ap[o]; a.y = Ap[o + 256];   // A^T frag: (m,kk) = A[kk][m]
            b.x = Bp[o]; b.y = Bp[o + 256];
            acc = __builtin_amdgcn_wmma_f32_16x16x4_f32(
                false, a, false, b, (short)0, acc, false, false);
        }
    }
#pragma unroll
    for (int v = 0; v < 8; ++v)
        C[(size_t)(m0 + v + lhalf * 8) * 256 + n0 + lmod] = acc[v];
}

// ---------------------------------------------------------------------------
// In-place Gauss-Jordan inversion of (I + G), G 256x256 SPD, entirely in LDS
// (257 KB dynamic LDS <= 320 KB/WGP). One column per thread.
// ---------------------------------------------------------------------------
__global__ void __launch_bounds__(256) gj_inverse_256(
    const float* __restrict__ G, float* __restrict__ Inv)
{
    extern __shared__ float sm[];
    float* Am   = sm;              // 256*256
    float* colk = sm + 256 * 256;  // 256
    const int j = threadIdx.x;

    for (int idx = j; idx < 256 * 256; idx += 256) {
        const int r = idx >> 8, c = idx & 255;
        Am[idx] = G[idx] + ((r == c) ? 1.0f : 0.0f);
    }
    __syncthreads();

    for (int k = 0; k < 256; ++k) {
        colk[j] = Am[j * 256 + k];          // snapshot pivot column
        __syncthreads();
        const float pivinv = 1.0f / colk[k];
        if (j == k) {                        // pivot column -> -f*pivinv, diag -> pivinv
#pragma unroll 4
            for (int i = 0; i < 256; ++i)
                Am[i * 256 + k] = (i == k) ? pivinv : -colk[i] * pivinv;
        } else {
            const float akj = Am[k * 256 + j] * pivinv;
            Am[k * 256 + j] = akj;
#pragma unroll 4
            for (int i = 0; i < 256; ++i)
                if (i != k) Am[i * 256 + j] -= colk[i] * akj;
        }
        __syncthreads();
    }
    for (int idx = j; idx < 256 * 256; idx += 256) Inv[idx] = Am[idx];
}

// ---------------------------------------------------------------------------
static inline void launch_gemm(hipStream_t s, const float* A, const float* B, float* C,
                               const float* Cadd, const float* bias,
                               int M, int N, int K, int lda, int ldb, int ldc,
                               float alpha, int mode)
{
    dim3 grid((M + 127) / 128, N / 128);
    gemm128<<<grid, 256, 0, s>>>(A, B, C, Cadd, bias, M, N, K, lda, ldb, ldc, alpha, mode);
}

extern "C" void kernel_launch(void* const* d_in, const int* in_sizes, int n_in,
                              void* d_out, int out_size, void* d_ws, size_t ws_size,
                              hipStream_t stream)
{
    (void)in_sizes; (void)n_in; (void)out_size; (void)ws_size;
    const float* features = (const float*)d_in[0];
    const float* distance = (const float*)d_in[1];
    const float* A_pa     = (const float*)d_in[2];
    const float* A_pp     = (const float*)d_in[3];
    const float* A_ap     = (const float*)d_in[4];
    const float* W0_pa    = (const float*)d_in[5];
    const float* W0_pp    = (const float*)d_in[6];
    const float* W0_ap    = (const float*)d_in[7];
    const float* W1_pa    = (const float*)d_in[8];
    const float* W1_pp    = (const float*)d_in[9];
    // d_in[10] W1_ap, d_in[13] fc_a_W, d_in[14] fc_a_b: dead code (embs2_a deleted)
    const float* fc_p_W   = (const float*)d_in[11];
    const float* fc_p_b   = (const float*)d_in[12];
    const float* mlp_W    = (const float*)d_in[15];
    const float* mlp_b    = (const float*)d_in[16];

    float* out_het = (float*)d_out;                               // embs_het 8000x256
    float* out_hom = (float*)d_out + (size_t)NPb * OUTb;          // embs_hom 8000x256

    float* ws   = (float*)d_ws;
    float* bufA = ws;                                // 8000x512
    float* bufB = bufA + (size_t)NPb * FTb;          // 8000x512
    float* bufC = bufB + (size_t)NPb * FTb;          // 8000x512
    float* bufE = bufC + (size_t)NPb * FTb;          // 8000x256
    float* bufF = bufE + (size_t)NPb * OUTb;         // 8000x256 (emb_f)
    float* bufG = bufF + (size_t)NPb * OUTb;         // 256x256
    float* bufH = bufG + 256 * 256;                  // 256x256 (inverse)

    const float* feat_p = features;
    const float* feat_a = features + (size_t)NPb * FTb;

    // ---- heterogeneous branch ----
    // embs1_a = relu(A_ap @ (feat_p @ W0_ap))
    launch_gemm(stream, feat_p, W0_ap, bufA, nullptr, nullptr, NPb, FTb, FTb, FTb, FTb, FTb, 1.f, MODE_PLAIN);
    launch_gemm(stream, A_ap, bufA, bufB, nullptr, nullptr, NPb, FTb, NPb, NPb, FTb, FTb, 1.f, MODE_RELU);
    // v_pa = relu(A_pa @ (feat_a @ W0_pa))
    launch_gemm(stream, feat_a, W0_pa, bufA, nullptr, nullptr, NPb, FTb, FTb, FTb, FTb, FTb, 1.f, MODE_PLAIN);
    launch_gemm(stream, A_pa, bufA, bufC, nullptr, nullptr, NPb, FTb, NPb, NPb, FTb, FTb, 1.f, MODE_RELU);
    // embs1_p = 0.5*(relu(A_pp @ (feat_p @ W0_pp)) + v_pa)   (fused epilogue, in-place on bufC)
    launch_gemm(stream, feat_p, W0_pp, bufA, nullptr, nullptr, NPb, FTb, FTb, FTb, FTb, FTb, 1.f, MODE_PLAIN);
    launch_gemm(stream, A_pp, bufA, bufC, bufC, nullptr, NPb, FTb, NPb, NPb, FTb, FTb, 1.f, MODE_AVG_RELU);
    // u_pa = relu(A_pa @ (embs1_a @ W1_pa))   (overwrites embs1_a)
    launch_gemm(stream, bufB, W1_pa, bufA, nullptr, nullptr, NPb, FTb, FTb, FTb, FTb, FTb, 1.f, MODE_PLAIN);
    launch_gemm(stream, A_pa, bufA, bufB, nullptr, nullptr, NPb, FTb, NPb, NPb, FTb, FTb, 1.f, MODE_RELU);
    // s_p = 0.5*(relu(A_pp @ (embs1_p @ W1_pp)) + u_pa)
    launch_gemm(stream, bufC, W1_pp, bufA, nullptr, nullptr, NPb, FTb, FTb, FTb, FTb, FTb, 1.f, MODE_PLAIN);
    launch_gemm(stream, A_pp, bufA, bufB, bufB, nullptr, NPb, FTb, NPb, NPb, FTb, FTb, 1.f, MODE_AVG_RELU);
    // embs_het = s_p @ fc_p_W[0:512] + feat_p @ fc_p_W[512:1024] + fc_p_b
    launch_gemm(stream, bufB, fc_p_W, bufE, nullptr, nullptr, NPb, OUTb, FTb, FTb, OUTb, OUTb, 1.f, MODE_PLAIN);
    launch_gemm(stream, feat_p, fc_p_W + (size_t)FTb * OUTb, out_het, bufE, fc_p_b,
                NPb, OUTb, FTb, FTb, OUTb, OUTb, 1.f, MODE_PLAIN);

    // ---- homophily branch ----
    // emb_f = feat_p @ mlp_W + mlp_b
    launch_gemm(stream, feat_p, mlp_W, bufF, nullptr, mlp_b, NPb, OUTb, FTb, FTb, OUTb, OUTb, 1.f, MODE_PLAIN);
    // G = emb_f^T @ emb_f
    gemm_atb256<<<dim3(16, 16), 32, 0, stream>>>(bufF, bufF, bufG, NPb);
    // inv = (I + G)^-1    (res = emb_f @ inv, since I - inv@G == inv with BETA=1)
    const int gj_smem = (256 * 256 + 256) * (int)sizeof(float);
    hipFuncSetAttribute(reinterpret_cast<const void*>(gj_inverse_256),
                        hipFuncAttributeMaxDynamicSharedMemorySize, gj_smem);
    gj_inverse_256<<<1, 256, gj_smem, stream>>>(bufG, bufH);
    // res = emb_f @ inv
    launch_gemm(stream, bufF, bufH, bufE, nullptr, nullptr, NPb, OUTb, OUTb, OUTb, OUTb, OUTb, 1.f, MODE_PLAIN);
    // tmp = emb_f^T @ res
    gemm_atb256<<<dim3(16, 16), 32, 0, stream>>>(bufF, bufE, bufG, NPb);
    // part1 = emb_f @ tmp
    launch_gemm(stream, bufF, bufG, bufC, nullptr, nullptr, NPb, OUTb, OUTb, OUTb, OUTb, OUTb, 1.f, MODE_PLAIN);
    // embs_hom = -0.5 * (distance @ res) + part1
    launch_gemm(stream, distance, bufE, out_hom, bufC, nullptr, NPb, OUTb, NPb, NPb, OUTb, OUTb, -0.5f, MODE_PLAIN);
}